// CorrNetImpl_66383014527325
// MI455X (gfx1250) — compile-verified
//
#include <hip/hip_runtime.h>
#include <cstdint>

#define Bb 8
#define Cc 64
#define Tt 32
#define Hh 56
#define Ww 56
#define HW 3136            // 56*56
#define CSTR 100352        // T*H*W (channel stride in x)
#define BSTR 6422528       // C*T*H*W (batch stride in x)
#define TH 8               // output rows per block
#define CCH 4              // channels staged per chunk
#define NCH (Cc / CCH)     // 16 chunks
#define X2R 14             // TH + 6 halo rows
#define X2W 64             // padded LDS row stride (interior at col 4..59)
#define NTHREADS 224       // 4 row-pairs * 56 columns

#if __has_builtin(__builtin_amdgcn_tensor_load_to_lds)
#define USE_TDM 1
#define AOPS 4
#else
#define USE_TDM 0
#define AOPS 6
#endif

typedef unsigned int v4u __attribute__((ext_vector_type(4)));
typedef int v8i __attribute__((ext_vector_type(8)));
typedef int v4i __attribute__((ext_vector_type(4)));
typedef float v2f __attribute__((ext_vector_type(2)));

__device__ __forceinline__ void async_b128(uint32_t lds_off, uint64_t gaddr) {
  // CDNA5 async global->LDS DMA (per-lane), tracked by ASYNCcnt
  asm volatile("global_load_async_to_lds_b128 %0, %1, off"
               :: "v"(lds_off), "v"(gaddr) : "memory");
}

// Forced VOP3P packed fp32 math (2 MACs per instruction)
__device__ __forceinline__ v2f pk_mul(v2f a, v2f b) {
  v2f d;
  asm("v_pk_mul_f32 %0, %1, %2" : "=v"(d) : "v"(a), "v"(b));
  return d;
}
__device__ __forceinline__ void pk_fma_acc_s(v2f& acc, v2f w_sgpr, v2f p) {
  // weight pair lives in an even-aligned SGPR pair (block-uniform)
  asm("v_pk_fma_f32 %0, %1, %2, %0" : "+v"(acc) : "s"(w_sgpr), "v"(p));
}

__global__ __launch_bounds__(NTHREADS)
void corrw_kernel(const float* __restrict__ x,
                  const float* __restrict__ wfil,
                  float* __restrict__ out) {
  const int htile = blockIdx.x;          // 0..6
  const int t     = blockIdx.y;          // 0..31
  const int b     = blockIdx.z;          // 0..7
  const int h0    = htile * TH;
  const int t1    = (t == 0) ? 0 : (t - 1);
  const int tid   = (int)threadIdx.x;    // 0..223
  const int w     = tid % Ww;
  const int ry    = tid / Ww;            // 0..3 (row pair)

  __shared__ __align__(16) float x1s[2][CCH * TH * Ww];
  __shared__ __align__(16) float x2s[2][CCH * X2R * X2W];
  __shared__ __align__(16) float dummy[4];

  // Zero the x2 staging tiles once; halo columns/rows stay zero forever
  for (int i = tid; i < 2 * CCH * X2R * X2W; i += NTHREADS) (&x2s[0][0])[i] = 0.0f;
  __syncthreads();

  const uint64_t xb2 = (uint64_t)(uintptr_t)(x + (size_t)b * BSTR + (size_t)t  * HW);
  const uint64_t xb1 = (uint64_t)(uintptr_t)(x + (size_t)b * BSTR + (size_t)t1 * HW
                                               + (size_t)h0 * Ww);
  const uint32_t dummy_off = (uint32_t)(uintptr_t)&dummy[0];

  auto stage = [&](int ci, int buf) {
    const int c0 = ci * CCH;
    const uint32_t l2 = (uint32_t)(uintptr_t)&x2s[buf][0];
    // x2 tile: CCH channels x 14 rows x 56 valid cols, b128 per lane, padded
    // with dummy issues so every wave's ASYNCcnt advances by exactly AOPS.
    #pragma unroll
    for (int k = 0; k < 4; ++k) {
      int u   = tid + k * NTHREADS;        // 0..895; valid < 784
      int ch  = u / (X2R * 14);
      int rem = u - ch * (X2R * 14);
      int row = rem / 14;
      int seg = rem - row * 14;
      int g   = h0 - 3 + row;              // global row
      uint64_t ga = xb2 + 4ull * ((size_t)(unsigned)(c0 + ch) * CSTR
                                  + (size_t)(unsigned)g * Ww + (unsigned)(seg * 4));
      uint32_t lo = l2 + 4u * (uint32_t)((ch * X2R + row) * X2W + 4 + seg * 4);
      if (u >= CCH * X2R * 14 || g < 0 || g >= Hh) {
        ga = (uint64_t)(uintptr_t)x;       // safe dummy source
        lo = dummy_off;
      }
      async_b128(lo, ga);
    }
#if USE_TDM
    // x1 tile via Tensor Data Mover: 3D tile 56 x TH x CCH, contiguous in LDS
    if (tid < 32) {
      uint64_t ga = xb1 + 4ull * ((size_t)c0 * CSTR);
      uint32_t la = (uint32_t)(uintptr_t)&x1s[buf][0];
      v4u g0 = { 1u,                                   // count=1, load, no gather
                 la,                                   // lds_addr
                 (uint32_t)ga,                         // global_addr[31:0]
                 ((uint32_t)(ga >> 32) & 0x01FFFFFFu) | (2u << 30) };  // addr hi | type=2
      v8i g1 = { (int)(2u << 16),                      // data_size = 4B
                 (int)(56u << 16),                     // tensor_dim0 lo16
                 (int)(56u << 16),                     // tensor_dim1 lo16
                 (int)(56u << 16),                     // tile_dim0 = 56
                 (int)(TH | (CCH << 16)),              // tile_dim1, tile_dim2
                 (int)Ww,                              // tensor_dim0_stride lo32
                 (int)((CSTR & 0xFFFF) << 16),         // tensor_dim1_stride lo16
                 (int)(CSTR >> 16) };                  // tensor_dim1_stride hi32
      v4i g2 = { Cc, 1, 0, 0 };                        // tensor_dim2, tensor_dim3
      v4i g3 = { 0, 0, 0, 0 };
      v8i g4 = { 0, 0, 0, 0, 0, 0, 0, 0 };             // unused 5th group (zeros)
      __builtin_amdgcn_tensor_load_to_lds(g0, g1, g2, g3, g4, 0);
    }
#else
    // Fallback: async-copy x1 tile (CCH x TH x 14 b128 units = 2/thread)
    #pragma unroll
    for (int k = 0; k < 2; ++k) {
      int u   = tid + k * NTHREADS;   // 0..447
      int ch  = u / (TH * 14);
      int rem = u - ch * (TH * 14);
      int row = rem / 14;
      int seg = rem - row * 14;
      uint64_t ga = xb1 + 4ull * ((size_t)(c0 + ch) * CSTR + (size_t)row * Ww + seg * 4);
      uint32_t lo = (uint32_t)(uintptr_t)&x1s[buf][0]
                  + 4u * (uint32_t)((ch * TH + row) * Ww + seg * 4);
      async_b128(lo, ga);
    }
#endif
  };

  // Accumulators: per pixel, 7 dy-rows x (3 packed dx-pairs + 1 scalar dx=6)
  v2f  acc0p[7][3], acc1p[7][3];
  float acc0s[7],   acc1s[7];
  #pragma unroll
  for (int r = 0; r < 7; ++r) {
    #pragma unroll
    for (int k = 0; k < 3; ++k) { acc0p[r][k] = (v2f)(0.0f); acc1p[r][k] = (v2f)(0.0f); }
    acc0s[r] = 0.0f; acc1s[r] = 0.0f;
  }

  auto compute = [&](int ci, int buf) {
    #pragma unroll 1
    for (int cc = 0; cc < CCH; ++cc) {
      // Block-uniform weights -> SMEM scalar loads; dx-adjacent pairs feed
      // the v_pk_fma SGPR-pair operand directly.
      const float* __restrict__ wp = wfil + ((size_t)(ci * CCH + cc) * Tt + t) * 49;
      v2f  w2[7][3];
      float ws[7];
      #pragma unroll
      for (int r = 0; r < 7; ++r) {
        #pragma unroll
        for (int k = 0; k < 3; ++k)
          w2[r][k] = (v2f){ wp[r * 7 + 2 * k], wp[r * 7 + 2 * k + 1] };
        ws[r] = wp[r * 7 + 6];
      }
      const float* x1p = &x1s[buf][cc * (TH * Ww)];
      const float* x2p = &x2s[buf][cc * (X2R * X2W)];
      float a0 = x1p[(2 * ry)     * Ww + w];
      float a1 = x1p[(2 * ry + 1) * Ww + w];
      v2f a0v = (v2f){ a0, a0 };
      v2f a1v = (v2f){ a1, a1 };
      #pragma unroll
      for (int rr = 0; rr < 8; ++rr) {
        // union row rr serves dy=rr for pixel0 and dy=rr-1 for pixel1
        const float* rp = &x2p[(2 * ry + rr) * X2W + (w + 1)];
        v2f s0 = (v2f){ rp[0], rp[1] };
        v2f s1 = (v2f){ rp[2], rp[3] };
        v2f s2 = (v2f){ rp[4], rp[5] };
        float s6 = rp[6];
        if (rr < 7) {
          pk_fma_acc_s(acc0p[rr][0], w2[rr][0], pk_mul(a0v, s0));
          pk_fma_acc_s(acc0p[rr][1], w2[rr][1], pk_mul(a0v, s1));
          pk_fma_acc_s(acc0p[rr][2], w2[rr][2], pk_mul(a0v, s2));
          acc0s[rr] = fmaf(ws[rr], a0 * s6, acc0s[rr]);
        }
        if (rr > 0) {
          const int r = rr - 1;
          pk_fma_acc_s(acc1p[r][0], w2[r][0], pk_mul(a1v, s0));
          pk_fma_acc_s(acc1p[r][1], w2[r][1], pk_mul(a1v, s1));
          pk_fma_acc_s(acc1p[r][2], w2[r][2], pk_mul(a1v, s2));
          acc1s[r] = fmaf(ws[r], a1 * s6, acc1s[r]);
        }
      }
    }
  };

  stage(0, 0);
  #pragma unroll 1
  for (int ci = 0; ci < NCH; ++ci) {
    if (ci + 1 < NCH) {
      stage(ci + 1, (ci + 1) & 1);
      // chunk ci's AOPS async ops complete in order before next chunk's
      asm volatile("s_wait_asynccnt %0" :: "i"(AOPS) : "memory");
#if USE_TDM
      __builtin_amdgcn_s_wait_tensorcnt(1);
#endif
    } else {
      asm volatile("s_wait_asynccnt 0" ::: "memory");
#if USE_TDM
      __builtin_amdgcn_s_wait_tensorcnt(0);
#endif
    }
    __syncthreads();
    compute(ci, ci & 1);
    __syncthreads();
  }

  // Unpack accumulators and store (coalesced along w)
  float acc0[49], acc1[49];
  #pragma unroll
  for (int r = 0; r < 7; ++r) {
    #pragma unroll
    for (int k = 0; k < 3; ++k) {
      acc0[r * 7 + 2 * k]     = acc0p[r][k].x;
      acc0[r * 7 + 2 * k + 1] = acc0p[r][k].y;
      acc1[r * 7 + 2 * k]     = acc1p[r][k].x;
      acc1[r * 7 + 2 * k + 1] = acc1p[r][k].y;
    }
    acc0[r * 7 + 6] = acc0s[r];
    acc1[r * 7 + 6] = acc1s[r];
  }

  const float invc = 1.0f / 64.0f;
  const int h_base = h0 + 2 * ry;
  #pragma unroll
  for (int p = 0; p < 49; ++p) {
    size_t o = ((size_t)(b * 49 + p) * Tt + t) * HW + (size_t)h_base * Ww + w;
    out[o]      = acc0[p] * invc;
    out[o + Ww] = acc1[p] * invc;
  }
}

extern "C" void kernel_launch(void* const* d_in, const int* in_sizes, int n_in,
                              void* d_out, int out_size, void* d_ws, size_t ws_size,
                              hipStream_t stream) {
  const float* x    = (const float*)d_in[0];
  const float* wfil = (const float*)d_in[1];
  float* out        = (float*)d_out;
  dim3 grid(Hh / TH, Tt, Bb);   // (7, 32, 8)
  corrw_kernel<<<grid, dim3(NTHREADS, 1, 1), 0, stream>>>(x, wfil, out);
}